// DiffusionStar_25331717111899
// MI455X (gfx1250) — compile-verified
//
#include <hip/hip_runtime.h>
#include <math.h>

#define Bsz 64
#define Dk  3072
#define Ntr 50000
#define KC  64
#define NSPLIT 25
#define KSPL (Ntr / NSPLIT)   // 2000

typedef __attribute__((ext_vector_type(2))) float v2f;
typedef __attribute__((ext_vector_type(4))) float f4;
typedef __attribute__((ext_vector_type(8))) float v8f;

// workspace layout (float offsets)
#define OFF_A     0
#define OFF_DENO  64
#define OFF_CX    128
#define OFF_CXH   192
#define OFF_XSQ   256
#define OFF_RMAX  320
#define OFF_RSUM  384
#define OFF_LOG   448
#define OFF_PART  (OFF_LOG + Bsz * Ntr)

// ---- CDNA5 async memory->LDS helpers (GLOBAL_LOAD_ASYNC_TO_LDS, ASYNCcnt) ----
// Generic pointers to __shared__ are {aperture_hi32, lds_offset_lo32}; the low
// 32 bits are the LDS byte address the async instruction's VDST expects.
__device__ __forceinline__ unsigned lds_off(const void* p) {
    return (unsigned)(unsigned long long)p;
}
__device__ __forceinline__ void async_ld_b128(const void* gptr, const void* lptr) {
    unsigned loff = lds_off(lptr);
    unsigned long long ga = (unsigned long long)gptr;
    asm volatile("global_load_async_to_lds_b128 %0, %1, off"
                 :: "v"(loff), "v"(ga) : "memory");
}
__device__ __forceinline__ void async_ld_b128_nt(const void* gptr, const void* lptr) {
    unsigned loff = lds_off(lptr);
    unsigned long long ga = (unsigned long long)gptr;
    asm volatile("global_load_async_to_lds_b128 %0, %1, off th:TH_LOAD_NT"
                 :: "v"(loff), "v"(ga) : "memory");
}
__device__ __forceinline__ void wait_async_10() { asm volatile("s_wait_asynccnt 10" ::: "memory"); }
__device__ __forceinline__ void wait_async_2()  { asm volatile("s_wait_asynccnt 2"  ::: "memory"); }
__device__ __forceinline__ void wait_async_0()  { asm volatile("s_wait_asynccnt 0"  ::: "memory"); }

// ---------------- kernel 0: per-row coefficients + ||x||^2 ----------------
__global__ __launch_bounds__(256) void k0_coef(const float* __restrict__ x,
                                               const float* __restrict__ acp,
                                               const int* __restrict__ t,
                                               float* __restrict__ ws) {
    int b = blockIdx.x;
    __shared__ float red[256];
    const float* row = x + (size_t)b * Dk;
    float s = 0.f;
    for (int i = threadIdx.x; i < Dk; i += 256) { float v = row[i]; s += v * v; }
    red[threadIdx.x] = s;
    __syncthreads();
    for (int off = 128; off > 0; off >>= 1) {
        if (threadIdx.x < off) red[threadIdx.x] += red[threadIdx.x + off];
        __syncthreads();
    }
    if (threadIdx.x == 0) {
        float ac  = acp[t[b]];
        float a   = sqrtf(ac);
        float om  = 1.f - ac;
        float isq = 1.f / sqrtf(om);
        ws[OFF_A    + b] = a;
        ws[OFF_DENO + b] = -0.5f / om;
        ws[OFF_CX   + b] = isq;
        ws[OFF_CXH  + b] = a * isq;
        ws[OFF_XSQ  + b] = red[0];
    }
}

// ---------------- kernel 1: logits via WMMA, double-buffered async staging ----------------
// grid: Ntr/16 blocks, 128 threads (4 waves). Wave w owns M-tile w; block owns 16 train rows.
__global__ __launch_bounds__(128) void k1_logits(const float* __restrict__ x,
                                                 const float* __restrict__ tr,
                                                 float* __restrict__ ws) {
    __shared__ float xs[2][64][KC + 4];   // stride 68: conflict-free fragment reads
    __shared__ float ts[2][16][KC + 4];
    __shared__ float la[64], ld[64], lq[64];
    __shared__ float part[128];
    __shared__ float trsq[16];

    const int tid = threadIdx.x;
    const int n0  = blockIdx.x * 16;
    if (tid < 64) { la[tid] = ws[OFF_A + tid]; ld[tid] = ws[OFF_DENO + tid]; lq[tid] = ws[OFF_XSQ + tid]; }

    const int wid = tid >> 5, lane = tid & 31;
    const int ln = lane & 15, half2 = (lane >> 4) << 1;   // K sub-offset per A/B layout

    const int xrow = tid >> 1, xc0 = (tid & 1) * 32;      // x staging map
    const int trow = tid >> 3, tc0 = (tid & 7) * 8;       // train staging map

    const float* xg0 = x  + (size_t)xrow * Dk + xc0;
    const float* tg0 = tr + (size_t)(n0 + trow) * Dk + tc0;

    // prologue: async-stage chunk 0 into buffer 0 (10 async instrs per wave)
    #pragma unroll
    for (int q = 0; q < 8; ++q)
        async_ld_b128(xg0 + 4 * q, &xs[0][xrow][xc0 + 4 * q]);
    async_ld_b128_nt(tg0,     &ts[0][trow][tc0]);
    async_ld_b128_nt(tg0 + 4, &ts[0][trow][tc0 + 4]);

    v8f c = {};
    float sq = 0.f;
    const int NCH = Dk / KC;   // 48

    for (int ci = 0; ci < NCH; ++ci) {
        const int p = ci & 1;
        if (ci + 1 < NCH) {
            // issue next chunk into the other buffer (safe: consumed last iter, barrier passed)
            const float* xg = xg0 + (ci + 1) * KC;
            const float* tg = tg0 + (ci + 1) * KC;
            #pragma unroll
            for (int q = 0; q < 8; ++q)
                async_ld_b128(xg + 4 * q, &xs[p ^ 1][xrow][xc0 + 4 * q]);
            async_ld_b128_nt(tg,     &ts[p ^ 1][trow][tc0]);
            async_ld_b128_nt(tg + 4, &ts[p ^ 1][trow][tc0 + 4]);
            wait_async_10();          // in-order completion: current chunk is done
        } else {
            wait_async_0();
        }
        __syncthreads();              // all waves' current-buffer data visible

        // squared train norms from the staged LDS tile
        {
            f4 v0 = *(const f4*)&ts[p][trow][tc0];
            f4 v1 = *(const f4*)&ts[p][trow][tc0 + 4];
            sq += v0.x*v0.x + v0.y*v0.y + v0.z*v0.z + v0.w*v0.w
                + v1.x*v1.x + v1.y*v1.y + v1.z*v1.z + v1.w*v1.w;
        }

        // 16 K-steps of v_wmma_f32_16x16x4_f32
        #pragma unroll
        for (int i = 0; i < 16; ++i) {
            int kk = i * 4 + half2;
            v2f A, Bv;
            A.x  = xs[p][wid * 16 + ln][kk];  A.y  = xs[p][wid * 16 + ln][kk + 1];
            Bv.x = ts[p][ln][kk];             Bv.y = ts[p][ln][kk + 1];
            c = __builtin_amdgcn_wmma_f32_16x16x4_f32(false, A, false, Bv,
                                                      (short)0, c, false, false);
        }
        __syncthreads();              // all waves done reading buffer p -> refillable
    }

    part[tid] = sq;
    __syncthreads();
    if (tid < 16) {
        float s = 0.f;
        for (int j = 0; j < 8; ++j) s += part[tid * 8 + j];
        trsq[tid] = s;
    }
    __syncthreads();

    // C layout: VGPR j -> M = j (lanes 0-15) / j+8 (lanes 16-31); N = lane&15
    float* lg = ws + OFF_LOG;
    const int mofs = (lane < 16) ? 0 : 8;
    const float tq = trsq[ln];
    #pragma unroll
    for (int j = 0; j < 8; ++j) {
        int m = wid * 16 + j + mofs;
        float am = la[m];
        float nv = ld[m] * (lq[m] - 2.f * am * c[j] + am * am * tq);
        lg[(size_t)m * Ntr + n0 + ln] = nv;
    }
}

// ---------------- kernel 2: per-row softmax stats ----------------
__global__ __launch_bounds__(256) void k2_stats(float* __restrict__ ws) {
    int b = blockIdx.x;
    __shared__ float red[256];
    const float* lg = ws + OFF_LOG + (size_t)b * Ntr;
    float m = -3.4e38f;
    for (int i = threadIdx.x; i < Ntr; i += 256) m = fmaxf(m, lg[i]);
    red[threadIdx.x] = m;
    __syncthreads();
    for (int off = 128; off > 0; off >>= 1) {
        if (threadIdx.x < off) red[threadIdx.x] = fmaxf(red[threadIdx.x], red[threadIdx.x + off]);
        __syncthreads();
    }
    float rmax = red[0];
    __syncthreads();
    float s = 0.f;
    for (int i = threadIdx.x; i < Ntr; i += 256) s += __expf(lg[i] - rmax);
    red[threadIdx.x] = s;
    __syncthreads();
    for (int off = 128; off > 0; off >>= 1) {
        if (threadIdx.x < off) red[threadIdx.x] += red[threadIdx.x + off];
        __syncthreads();
    }
    if (threadIdx.x == 0) { ws[OFF_RMAX + b] = rmax; ws[OFF_RSUM + b] = red[0]; }
}

// ---------------- kernel 3: partial = exp(logit-max) @ train (split-K, deterministic) ----------------
// grid: (Dk/64, NSPLIT), 128 threads. Wave w owns M-tile w, 4 D-subtiles of 16.
__global__ __launch_bounds__(128) void k3_wsum(const float* __restrict__ tr,
                                               float* __restrict__ ws) {
    __shared__ float wt[64][20];       // weights tile, stride 20 (conflict-free A reads)
    __shared__ float tt[2][16][72];    // train tile x2, stride 72 (conflict-free B reads)
    __shared__ float lrm[64];

    const int tid   = threadIdx.x;
    const int dbase = blockIdx.x * 64;
    const int spl   = blockIdx.y;
    const int k0s   = spl * KSPL;
    if (tid < 64) lrm[tid] = ws[OFF_RMAX + tid];

    const int wid = tid >> 5, lane = tid & 31;
    const int ln = lane & 15, half2 = (lane >> 4) << 1;
    const int wrow = tid >> 1, wc0 = (tid & 1) * 8;
    const int trow = tid >> 3, tc0 = (tid & 7) * 8;

    const float* lg  = ws + OFF_LOG;
    const float* tg0 = tr + (size_t)(k0s + trow) * Dk + dbase + tc0;

    // prologue: async-stage chunk 0 into buffer 0 (2 async instrs per wave)
    async_ld_b128_nt(tg0,     &tt[0][trow][tc0]);
    async_ld_b128_nt(tg0 + 4, &tt[0][trow][tc0 + 4]);
    __syncthreads();   // lrm visible before first weight staging

    v8f c0 = {}, c1 = {}, c2 = {}, c3 = {};
    const int NCH = KSPL / 16;   // 125

    for (int ci = 0; ci < NCH; ++ci) {
        const int p = ci & 1;
        if (ci + 1 < NCH) {
            const float* tg = tg0 + (size_t)(ci + 1) * 16 * Dk;
            async_ld_b128_nt(tg,     &tt[p ^ 1][trow][tc0]);
            async_ld_b128_nt(tg + 4, &tt[p ^ 1][trow][tc0 + 4]);
        }
        // stage softmax weights (unnormalized) for this chunk; overlaps async loads
        {
            const int k0 = k0s + ci * 16;
            const float* lp = lg + (size_t)wrow * Ntr + k0 + wc0;
            float rm = lrm[wrow];
            f4 v0 = *(const f4*)lp;
            f4 v1 = *(const f4*)(lp + 4);
            v0.x = __expf(v0.x - rm); v0.y = __expf(v0.y - rm);
            v0.z = __expf(v0.z - rm); v0.w = __expf(v0.w - rm);
            v1.x = __expf(v1.x - rm); v1.y = __expf(v1.y - rm);
            v1.z = __expf(v1.z - rm); v1.w = __expf(v1.w - rm);
            *(f4*)&wt[wrow][wc0]     = v0;
            *(f4*)&wt[wrow][wc0 + 4] = v1;
        }
        if (ci + 1 < NCH) wait_async_2(); else wait_async_0();
        __syncthreads();

        #pragma unroll
        for (int i = 0; i < 4; ++i) {
            int kk = i * 4 + half2;
            v2f A;
            A.x = wt[wid * 16 + ln][kk]; A.y = wt[wid * 16 + ln][kk + 1];
            v2f B0, B1, B2, B3;
            B0.x = tt[p][kk][ln];      B0.y = tt[p][kk + 1][ln];
            B1.x = tt[p][kk][16 + ln]; B1.y = tt[p][kk + 1][16 + ln];
            B2.x = tt[p][kk][32 + ln]; B2.y = tt[p][kk + 1][32 + ln];
            B3.x = tt[p][kk][48 + ln]; B3.y = tt[p][kk + 1][48 + ln];
            c0 = __builtin_amdgcn_wmma_f32_16x16x4_f32(false, A, false, B0, (short)0, c0, false, false);
            c1 = __builtin_amdgcn_wmma_f32_16x16x4_f32(false, A, false, B1, (short)0, c1, false, false);
            c2 = __builtin_amdgcn_wmma_f32_16x16x4_f32(false, A, false, B2, (short)0, c2, false, false);
            c3 = __builtin_amdgcn_wmma_f32_16x16x4_f32(false, A, false, B3, (short)0, c3, false, false);
        }
        __syncthreads();   // buffer p + wt fully consumed -> refillable next iter
    }

    // write this split's partial (disjoint per workgroup -> deterministic)
    float* outp = ws + OFF_PART + (size_t)spl * (Bsz * Dk);
    const int mofs = (lane < 16) ? 0 : 8;
    #pragma unroll
    for (int j = 0; j < 8; ++j) {
        int m = wid * 16 + j + mofs;
        float* r = outp + (size_t)m * Dk + dbase + ln;
        r[0]  = c0[j];
        r[16] = c1[j];
        r[32] = c2[j];
        r[48] = c3[j];
    }
}

// ---------------- kernel 4: reduce splits + final combine ----------------
__global__ __launch_bounds__(256) void k4_final(const float* __restrict__ x,
                                                const float* __restrict__ ws,
                                                float* __restrict__ out) {
    int idx = blockIdx.x * 256 + threadIdx.x;
    if (idx >= Bsz * Dk) return;
    int b = idx / Dk;
    const float* part = ws + OFF_PART;
    float s = 0.f;
    #pragma unroll
    for (int p = 0; p < NSPLIT; ++p) s += part[(size_t)p * (Bsz * Dk) + idx];
    float inv = 1.f / ws[OFF_RSUM + b];
    out[idx] = ws[OFF_CX + b] * x[idx] - ws[OFF_CXH + b] * s * inv;
}

extern "C" void kernel_launch(void* const* d_in, const int* in_sizes, int n_in,
                              void* d_out, int out_size, void* d_ws, size_t ws_size,
                              hipStream_t stream) {
    (void)in_sizes; (void)n_in; (void)out_size; (void)ws_size;
    const float* x   = (const float*)d_in[0];
    const float* tr  = (const float*)d_in[1];
    const float* acp = (const float*)d_in[2];
    const int*   t   = (const int*)d_in[3];
    float* ws  = (float*)d_ws;
    float* out = (float*)d_out;

    k0_coef <<<Bsz, 256, 0, stream>>>(x, acp, t, ws);
    k1_logits<<<Ntr / 16, 128, 0, stream>>>(x, tr, ws);
    k2_stats<<<Bsz, 256, 0, stream>>>(ws);
    k3_wsum <<<dim3(Dk / 64, NSPLIT), 128, 0, stream>>>(tr, ws);
    k4_final<<<(Bsz * Dk + 255) / 256, 256, 0, stream>>>(x, ws, out);
}